// Myloss_22110491639941
// MI455X (gfx1250) — compile-verified
//
#include <hip/hip_runtime.h>
#include <hip/hip_bf16.h>

typedef __attribute__((ext_vector_type(2))) float v2f;
typedef __attribute__((ext_vector_type(8))) float v8f;

#define MAXD 2048
#define SPAD 4                     // LDS row padding (floats) -> bank-conflict-free
#define SROW (MAXD + SPAD)         // 2052 % 64 == 4: distinct banks for 16 rows

// ---------------------------------------------------------------------------
// Kernel 1: 3x3 Gram of standar_score via V_WMMA_F32_16X16X4_F32.
// Stage S into LDS (bulk coalesced loads, 256 threads), zero-pad a 4th row,
// then one wave runs the WMMA loop from LDS with straight-line ds_load_b64
// (no divergence: lanes r>=3 read the zero row). Gram entries extracted with
// __shfl lane broadcasts (no barrier needed inside the wave-only region).
// cos = G_ab / sqrt(G_aa) * sqrt(G_bb)  =>  cos^2 = G_ab^2 * G_bb / G_aa.
// Writes d_out[0] (non-atomic) => also initializes the accumulator.
// ---------------------------------------------------------------------------
__global__ __launch_bounds__(256) void gram_cos_kernel(
    const float* __restrict__ S, float* __restrict__ out, int D) {
  __shared__ __align__(16) float sS[4 * SROW];   // rows 0..2 = S, row 3 = 0

  const int tid = threadIdx.x;
  const int d4 = D >> 2;

  // Stage 3 rows of S into padded LDS rows (b128 loads, one wait at barrier).
  for (int i = tid; i < 3 * d4; i += 256) {
    const int row = i / d4;
    const int col = i - row * d4;
    *(float4*)(sS + row * SROW + col * 4) = ((const float4*)S)[i];
  }
  // Zero row 3 (read by lanes whose matrix row is padding).
  for (int i = tid; i < d4; i += 256) {
    *(float4*)(sS + 3 * SROW + i * 4) = make_float4(0.f, 0.f, 0.f, 0.f);
  }
  __syncthreads();

  if (tid < 32) {                       // single wave32 does the WMMA loop
    const int r  = tid & 15;            // matrix row for this lane's slots
    const int hi = tid >> 4;            // lanes 16..31 carry K = k+2, k+3
    const float* base = sS + (r < 3 ? r : 3) * SROW + 2 * hi;

    v8f c = {0.f, 0.f, 0.f, 0.f, 0.f, 0.f, 0.f, 0.f};
    // A (16x4) and B (4x16) have identical lane/VGPR contents for S*S^T:
    //   lane<16 : slot0 = S[r][k+0], slot1 = S[r][k+1]
    //   lane>=16: slot0 = S[r][k+2], slot1 = S[r][k+3]
#pragma unroll 8
    for (int k = 0; k < D; k += 4) {
      v2f a = *(const v2f*)(base + k);  // ds_load_b64, bank-conflict-free
      c = __builtin_amdgcn_wmma_f32_16x16x4_f32(
          /*neg_a=*/false, a, /*neg_b=*/false, a,
          /*c_mod=*/(short)0, c, /*reuse_a=*/false, /*reuse_b=*/false);
    }

    // D layout: VGPR i, lanes 0..15 hold row M=i, N=lane.
    float g00 = __shfl(c[0], 0, 32);
    float g01 = __shfl(c[0], 1, 32);
    float g02 = __shfl(c[0], 2, 32);
    float g11 = __shfl(c[1], 1, 32);
    float g12 = __shfl(c[1], 2, 32);
    float g22 = __shfl(c[2], 2, 32);

    if (tid == 0) {
      float c01 = g01 * g01 * (g11 / g00);
      float c02 = g02 * g02 * (g22 / g00);
      float c12 = g12 * g12 * (g22 / g11);
      out[0] = c01 + c02 + c12;
    }
  }
}

// ---------------------------------------------------------------------------
// Kernel 2: loss_min = sum_b || x_b - S[polarity_b] ||^2.
// HBM-bound streaming reduction (~128 MiB @ 23.3 TB/s ~= 5.8 us floor):
// float4 loads of x, S cached in LDS, wave32 shuffle reduction, one
// global_atomic_add_f32 per block.
// ---------------------------------------------------------------------------
__global__ __launch_bounds__(256) void loss_min_kernel(
    const float* __restrict__ x, const int* __restrict__ pol,
    const float* __restrict__ S, float* __restrict__ out,
    int B, int D, int rowsPerBlock) {
  __shared__ __align__(16) float sS[3 * MAXD];

  const int tid = threadIdx.x;

  // Preload standar_score (3*D floats = 24 KB) into LDS with b128 loads.
  {
    const int nf4 = (3 * D) >> 2;
    const float4* S4 = (const float4*)S;
    float4* sS4 = (float4*)sS;
    for (int i = tid; i < nf4; i += (int)blockDim.x) sS4[i] = S4[i];
  }
  __syncthreads();

  const int d4 = D >> 2;                       // float4s per row (512 for D=2048)
  const int row0 = blockIdx.x * rowsPerBlock;
  const int rowEnd = min(row0 + rowsPerBlock, B);

  float acc = 0.f;
  for (int b = row0; b < rowEnd; ++b) {
    const int p = pol[b];                      // uniform -> scalar load
    const float4* xr = (const float4*)(x + (size_t)b * (size_t)D);
    const float4* sr = (const float4*)(sS + p * D);
#pragma unroll 2
    for (int i = tid; i < d4; i += 256) {      // 2 b128 global loads/row/thread
      float4 xv = xr[i];
      float4 sv = sr[i];
      float dx = xv.x - sv.x;
      float dy = xv.y - sv.y;
      float dz = xv.z - sv.z;
      float dw = xv.w - sv.w;
      acc = fmaf(dx, dx, acc);
      acc = fmaf(dy, dy, acc);
      acc = fmaf(dz, dz, acc);
      acc = fmaf(dw, dw, acc);
    }
  }

  // wave32 butterfly reduction
#pragma unroll
  for (int off = 16; off > 0; off >>= 1)
    acc += __shfl_xor(acc, off, 32);

  __shared__ float wsum[8];
  const int wid = tid >> 5;
  if ((tid & 31) == 0) wsum[wid] = acc;
  __syncthreads();

  if (tid == 0) {
    float s = 0.f;
#pragma unroll
    for (int w = 0; w < 8; ++w) s += wsum[w];
    atomicAdd(out, s);                         // global_atomic_add_f32
  }
}

// ---------------------------------------------------------------------------
extern "C" void kernel_launch(void* const* d_in, const int* in_sizes, int n_in,
                              void* d_out, int out_size, void* d_ws, size_t ws_size,
                              hipStream_t stream) {
  const float* x   = (const float*)d_in[0];   // [B, D] f32
  const int*   pol = (const int*)d_in[1];     // [B] int
  const float* S   = (const float*)d_in[2];   // [3, D] f32
  float* out = (float*)d_out;                 // scalar f32

  const int B = in_sizes[1];
  const int D = in_sizes[2] / 3;

  // Runs first on the stream: computes cosine terms AND initializes out[0].
  gram_cos_kernel<<<1, 256, 0, stream>>>(S, out, D);

  const int blocks = 512;                     // 12 MB of S re-reads (~9% of x)
  const int rpb = (B + blocks - 1) / blocks;  // 32 rows/block for B=16384
  loss_min_kernel<<<blocks, 256, 0, stream>>>(x, pol, S, out, B, D, rpb);
}